// Interpolator_54133767799514
// MI455X (gfx1250) — compile-verified
//
#include <hip/hip_runtime.h>

// Depthwise conv_transpose2d (bilinear up-4, k=7) on MI455X.
// Bandwidth-bound: ~292 MB traffic -> ~12.5us floor @ 23.3 TB/s.
// Strategy: TDM (tensor_load_to_lds) stages the input row-band per block into
// LDS (49x reuse per input pixel), compute is 4 taps/output, stores coalesced.

#define IH 64
#define IW 64
#define OH 259
#define OW 259
#define NCH 256
#define NBATCH 4
#define STRIPS 4
#define STRIP_ROWS 65     // 65+65+65+64 = 259
#define MAXROWS 18        // max staged input rows per strip (+halo)
#define KS 7

typedef unsigned int u32x4 __attribute__((ext_vector_type(4)));
typedef int          i32x8 __attribute__((ext_vector_type(8)));
typedef int          i32x4 __attribute__((ext_vector_type(4)));

__global__ __launch_bounds__(256)
void Interpolator_54133767799514_kernel(const float* __restrict__ x,
                                        const float* __restrict__ w,
                                        float* __restrict__ out)
{
    __shared__ float smx[MAXROWS * IW];   // staged input tile
    __shared__ float smw[KS * KS];        // this channel's 7x7 weights

    const int strip = blockIdx.x & 3;
    const int plane = blockIdx.x >> 2;          // n*NCH + c
    const int chan  = plane & (NCH - 1);

    const int oy0 = strip * STRIP_ROWS;
    const int oyE = (oy0 + STRIP_ROWS < OH) ? (oy0 + STRIP_ROWS) : OH;

    // input rows needed: jy-1 .. jy for jy in [oy0>>2, (oyE-1)>>2], clipped
    const int iy_lo = ((oy0 >> 2) - 1) > 0 ? ((oy0 >> 2) - 1) : 0;
    int iy_hi = (oyE - 1) >> 2;
    if (iy_hi > IH - 1) iy_hi = IH - 1;
    const int nrows = iy_hi - iy_lo + 1;        // <= MAXROWS

    const float* gsrc = x + ((size_t)plane * (IH * IW) + (size_t)iy_lo * IW);

    // stage per-channel weights (tiny, 196B, L2-resident)
    if (threadIdx.x < KS * KS)
        smw[threadIdx.x] = w[chan * (KS * KS) + threadIdx.x];

#if __has_builtin(__builtin_amdgcn_tensor_load_to_lds)
    // ---- Tensor Data Mover: DMA nrows x 64 f32 tile global -> LDS ----
    if (threadIdx.x < 32) {   // one wave issues the descriptor (EXEC ignored by TDM)
        // generic shared ptr low 32 bits == LDS byte offset (flat LDS aperture rule)
        unsigned lds_addr = (unsigned)(unsigned long long)(&smx[0]);
        unsigned long long ga = (unsigned long long)gsrc;

        u32x4 g0;
        g0[0] = 1u;                                   // count=1, user mode, no gather
        g0[1] = lds_addr;                             // lds_addr [63:32]
        g0[2] = (unsigned)ga;                         // global_addr [95:64]
        g0[3] = (unsigned)((ga >> 32) & 0x01FFFFFFu)  // global_addr [120:96]
              | (2u << 30);                           // type=2 ("image")

        i32x8 g1;
        g1[0] = (2 << 16);        // data_size=2 (4B); mask/pad/iterate = 0
        g1[1] = (IW << 16);       // tensor_dim0 = 64 (low16; high16 in g1[2][15:0]=0)
        g1[2] = (nrows << 16);    // tensor_dim1 = nrows (low16)
        g1[3] = (IW << 16);       // tile_dim0 = 64
        g1[4] = nrows;            // tile_dim1 = nrows; tile_dim2 = 0 (2-D)
        g1[5] = IW;               // tensor_dim0_stride = 64 (low32)
        g1[6] = 0;                // stride0 hi | stride1 lo
        g1[7] = 0;                // stride1 hi

        i32x4 gz4 = {0, 0, 0, 0};           // groups 2/3 unused for a 2-D tensor
        i32x8 gz8 = {0, 0, 0, 0, 0, 0, 0, 0};  // clang-23 6-arg form trailing group
        __builtin_amdgcn_tensor_load_to_lds(g0, g1, gz4, gz4, gz8, 0);
        __builtin_amdgcn_s_wait_tensorcnt(0);
    }
#else
    // fallback: cooperative load (rows are contiguous: stride == tile width)
    for (int i = threadIdx.x; i < nrows * IW; i += 256) smx[i] = gsrc[i];
#endif
    __syncthreads();

    // ---- x-side tap math is row-invariant: hoist it per thread ----
    // slot 0: ox = tid (covers 0..255); slot 1: ox = tid+256 (only tid<3)
    int   kxa[2], kxb[2], ca[2], cb[2], oxv[2], nslot = 1;
    float mxa[2], mxb[2];
    {
        oxv[0] = threadIdx.x;
        oxv[1] = threadIdx.x + 256;
        if (oxv[1] < OW) nslot = 2;
        #pragma unroll
        for (int s = 0; s < 2; ++s) {
            const int ox = oxv[s];
            const int px = ox & 3, jx = ox >> 2;
            kxa[s] = px;
            kxb[s] = (px + 4 < KS) ? (px + 4) : (KS - 1);
            mxa[s] = (jx <= IW - 1) ? 1.f : 0.f;
            mxb[s] = ((px < 3) && (jx >= 1)) ? 1.f : 0.f;
            ca[s]  = (jx <= IW - 1) ? jx : (IW - 1);
            cb[s]  = (jx >= 1) ? (jx - 1) : 0;
        }
    }

    float* oplane = out + (size_t)plane * (OH * OW);

    for (int oy = oy0; oy < oyE; ++oy) {
        const int py  = oy & 3, jy = oy >> 2;
        const int kya = py;
        const int kyb = (py + 4 < KS) ? (py + 4) : (KS - 1);
        const float mya = (jy <= IH - 1) ? 1.f : 0.f;
        const float myb = ((py < 3) && (jy >= 1)) ? 1.f : 0.f;
        const int ra = (((jy <= IH - 1) ? jy : (IH - 1)) - iy_lo) * IW;
        const int rb = (((jy >= 1) ? (jy - 1) : 0)        - iy_lo) * IW;
        float* orow = oplane + (size_t)oy * OW;

        #pragma unroll
        for (int s = 0; s < 2; ++s) {
            if (s >= nslot) break;
            const float wAA = smw[kya * KS + kxa[s]] * (mya * mxa[s]);
            const float wAB = smw[kya * KS + kxb[s]] * (mya * mxb[s]);
            const float wBA = smw[kyb * KS + kxa[s]] * (myb * mxa[s]);
            const float wBB = smw[kyb * KS + kxb[s]] * (myb * mxb[s]);
            const float v = wAA * smx[ra + ca[s]] + wAB * smx[ra + cb[s]]
                          + wBA * smx[rb + ca[s]] + wBB * smx[rb + cb[s]];
            orow[oxv[s]] = v;
        }
    }
}

extern "C" void kernel_launch(void* const* d_in, const int* in_sizes, int n_in,
                              void* d_out, int out_size, void* d_ws, size_t ws_size,
                              hipStream_t stream) {
    (void)in_sizes; (void)n_in; (void)d_ws; (void)ws_size; (void)out_size;
    const float* x   = (const float*)d_in[0];  // [4,256,64,64] f32
    const float* wgt = (const float*)d_in[1];  // [256,1,7,7]  f32
    float* out = (float*)d_out;                // [4,256,259,259] f32

    dim3 grid(NBATCH * NCH * STRIPS);          // 4096 blocks
    Interpolator_54133767799514_kernel<<<grid, 256, 0, stream>>>(x, wgt, out);
}